// SimilarityBasedRelationEnhancer_27015344291947
// MI455X (gfx1250) — compile-verified
//
#include <hip/hip_runtime.h>
#include <hip/hip_bf16.h>

#define Bn 128
#define Rn 4096
#define Dn 256
#define TILE_ROWS 128
#define LDS_STRIDE 260   // 260*4 = 1040 bytes: 16B-aligned rows, bank stride 4 -> conflict-free A fetch

typedef __attribute__((ext_vector_type(2))) float v2f;
typedef __attribute__((ext_vector_type(8))) float v8f;

__device__ __forceinline__ float sigmoidf_(float x) { return 1.0f / (1.0f + __expf(-x)); }

// ---------------------------------------------------------------------------
// K1: per-sample query normalization + zero the weighted-sum accumulator
// ---------------------------------------------------------------------------
__global__ __launch_bounds__(256) void prep_kernel(const float* __restrict__ reps,
                                                   const int* __restrict__ qidx,
                                                   float* __restrict__ qn_ws,
                                                   float* __restrict__ acc_ws) {
  __shared__ float red[256];
  const int b = blockIdx.x, t = threadIdx.x;
  const int qi = qidx[b];
  const float qv = reps[(long long)b * Rn * Dn + (long long)qi * Dn + t];
  red[t] = qv * qv;
  __syncthreads();
  for (int s = 128; s > 0; s >>= 1) {
    if (t < s) red[t] += red[t + s];
    __syncthreads();
  }
  const float inv = 1.0f / fmaxf(sqrtf(red[0]), 1e-12f);
  qn_ws[b * Dn + t] = qv * inv;
  acc_ws[b * Dn + t] = 0.0f;
}

// ---------------------------------------------------------------------------
// K2: fused stream pass. Per (b, 128-row tile):
//   - copy rows to out while staging in LDS (coalesced float4)
//   - per-row L2 norms from LDS
//   - sims via V_WMMA_F32_16X16X4_F32: A = 16 rows x 4 K-chunk, B = qn chunk
//     broadcast into all 16 columns -> D[m][*] = dot(row m, qn)
// ---------------------------------------------------------------------------
__global__ __launch_bounds__(256) void sims_kernel(const float* __restrict__ reps,
                                                   const int* __restrict__ qidx,
                                                   const float* __restrict__ qn_ws,
                                                   float* __restrict__ out,
                                                   float* __restrict__ sims_ws) {
  __shared__ float tile[TILE_ROWS * LDS_STRIDE];
  __shared__ float qsh[Dn];
  __shared__ float nrm2[TILE_ROWS * 2];
  __shared__ float invn[TILE_ROWS];

  const int b = blockIdx.y;
  const int rowbase = blockIdx.x * TILE_ROWS;
  const int t = threadIdx.x;
  const long long gbase = (long long)b * Rn * Dn + (long long)rowbase * Dn;

  if (t < Dn) qsh[t] = qn_ws[b * Dn + t];

  // stage + copy-out: 256 threads x float4 = 4 rows per iteration
  const int r0 = t >> 6;            // 0..3
  const int c4 = (t & 63) << 2;     // 0..252
  for (int it = 0; it < TILE_ROWS / 4; ++it) {
    const int r = it * 4 + r0;
    const float4 v = *(const float4*)(reps + gbase + (long long)r * Dn + c4);
    *(float4*)(out + gbase + (long long)r * Dn + c4) = v;
    *(float4*)(&tile[r * LDS_STRIDE + c4]) = v;
  }
  __syncthreads();

  // row norms: 2 threads per row, 128 elements each
  {
    const int r = t >> 1, h = t & 1;
    const float* p = &tile[r * LDS_STRIDE + h * 128];
    float s = 0.0f;
#pragma unroll 8
    for (int j = 0; j < 128; ++j) s += p[j] * p[j];
    nrm2[r * 2 + h] = s;
  }
  __syncthreads();
  if (t < TILE_ROWS) invn[t] = 1.0f / fmaxf(sqrtf(nrm2[t * 2] + nrm2[t * 2 + 1]), 1e-12f);
  __syncthreads();

  // WMMA GEMV: wave w owns rows [w*16, w*16+16)
  const int wave = t >> 5;
  const int lane = t & 31;
  const int mrow = lane & 15;
  const int ksel = (lane >> 4) << 1;   // lanes 0-15 -> K 0,1 ; lanes 16-31 -> K 2,3
  const float* arow = &tile[(wave * 16 + mrow) * LDS_STRIDE];

  v8f c = {0.0f, 0.0f, 0.0f, 0.0f, 0.0f, 0.0f, 0.0f, 0.0f};
#pragma unroll 4
  for (int kb = 0; kb < Dn; kb += 4) {
    const int kk = kb + ksel;
    const v2f a  = *(const v2f*)(arow + kk);   // A[m][k] = row element
    const v2f bb = *(const v2f*)(&qsh[kk]);    // B[k][n] = qn[k] for every column n
    c = __builtin_amdgcn_wmma_f32_16x16x4_f32(false, a, false, bb, (short)0, c,
                                              false, false);
  }

  // D layout: VGPR j holds M=j (lanes 0-15) and M=j+8 (lanes 16-31); any column works
  const int qi = qidx[b];
  if (lane == 0 || lane == 16) {
    const int moff = (lane == 0) ? 0 : 8;
#pragma unroll
    for (int j = 0; j < 8; ++j) {
      const int rl = wave * 16 + moff + j;
      const int rg = rowbase + rl;
      float s = c[j] * invn[rl];               // dot(r_i, qn) / max(||r_i||, eps)
      if (rg == qi) s = -1.0f;                 // exclude self
      sims_ws[(long long)b * Rn + rg] = s;
    }
  }
}

// ---------------------------------------------------------------------------
// K3: per-sample gated softmax -> adjusted weights (in-place over sims)
// ---------------------------------------------------------------------------
__global__ __launch_bounds__(256) void weights_kernel(float* __restrict__ sims_ws,
                                                      const float* __restrict__ thr_raw,
                                                      const float* __restrict__ wscale_p,
                                                      const float* __restrict__ temp_p,
                                                      float* __restrict__ flag_ws) {
  __shared__ float red[256];
  const int b = blockIdx.x, t = threadIdx.x;
  const float thr = sigmoidf_(thr_raw[0]);
  const float temp = fminf(fmaxf(temp_p[0], 0.1f), 10.0f);
  const float wscale = wscale_p[0];
  float* s = sims_ws + (long long)b * Rn;

  // pass 1: max logit (masked -> -1e9; valid logits are >= -10)
  float lmax = -1e9f;
  for (int i = t; i < Rn; i += 256) {
    const float si = s[i];
    lmax = fmaxf(lmax, (si > thr) ? (si / temp) : -1e9f);
  }
  red[t] = lmax;
  __syncthreads();
  for (int k = 128; k > 0; k >>= 1) { if (t < k) red[t] = fmaxf(red[t], red[t + k]); __syncthreads(); }
  const float gmax = red[0];
  __syncthreads();

  // pass 2: sum of exp
  float se = 0.0f;
  for (int i = t; i < Rn; i += 256) {
    const float si = s[i];
    se += __expf(((si > thr) ? (si / temp) : -1e9f) - gmax);
  }
  red[t] = se;
  __syncthreads();
  for (int k = 128; k > 0; k >>= 1) { if (t < k) red[t] += red[t + k]; __syncthreads(); }
  const float gse = red[0];
  __syncthreads();

  // pass 3: unnormalized adjusted weights + their sum (thread owns its indices)
  float asum = 0.0f;
  for (int i = t; i < Rn; i += 256) {
    const float si = s[i];
    const bool m = si > thr;
    const float w = __expf((m ? (si / temp) : -1e9f) - gmax) / gse;
    const float simw = sigmoidf_((si - thr) * 10.0f);
    const float adj = m ? (w * simw * (1.0f + wscale * si)) : 0.0f;
    s[i] = adj;
    asum += adj;
  }
  red[t] = asum;
  __syncthreads();
  for (int k = 128; k > 0; k >>= 1) { if (t < k) red[t] += red[t + k]; __syncthreads(); }
  const float scale = 1.0f / (red[0] + 1e-8f);
  __syncthreads();

  for (int i = t; i < Rn; i += 256) s[i] *= scale;
  if (t == 0) flag_ws[b] = (gmax > -0.5e9f) ? 1.0f : 0.0f;   // any row passed the gate?
}

// ---------------------------------------------------------------------------
// K4: sparse weighted row-sum: only touch rows with nonzero adjusted weight
// ---------------------------------------------------------------------------
__global__ __launch_bounds__(256) void wsum_kernel(const float* __restrict__ reps,
                                                   const float* __restrict__ adj_ws,
                                                   float* __restrict__ acc_ws) {
  __shared__ float adjsh[256];
  const int b = blockIdx.y;
  const int rowbase = blockIdx.x * 256;
  const int t = threadIdx.x;
  adjsh[t] = adj_ws[(long long)b * Rn + rowbase + t];
  __syncthreads();

  const float* base = reps + (long long)b * Rn * Dn + (long long)rowbase * Dn;
  float acc = 0.0f;
  bool any = false;
  for (int r = 0; r < 256; ++r) {
    const float a = adjsh[r];              // wave-uniform: skip is branch-cheap
    if (a != 0.0f) {
      acc = fmaf(a, base[(long long)r * Dn + t], acc);
      any = true;
    }
  }
  if (any) atomicAdd(&acc_ws[b * Dn + t], acc);
}

// ---------------------------------------------------------------------------
// K5: blend and write the replaced row
// ---------------------------------------------------------------------------
__global__ __launch_bounds__(256) void finalize_kernel(const float* __restrict__ reps,
                                                       const int* __restrict__ qidx,
                                                       const float* __restrict__ acc_ws,
                                                       const float* __restrict__ flag_ws,
                                                       const float* __restrict__ str_raw,
                                                       float* __restrict__ out) {
  const int b = blockIdx.x, t = threadIdx.x;
  const int qi = qidx[b];
  const float strength = 0.2f * sigmoidf_(str_raw[0]);
  const float q = reps[(long long)b * Rn * Dn + (long long)qi * Dn + t];
  const float wv = acc_ws[b * Dn + t];
  const float nv = (flag_ws[b] > 0.5f) ? ((1.0f - strength) * q + strength * wv) : q;
  out[(long long)b * Rn * Dn + (long long)qi * Dn + t] = nv;
}

// ---------------------------------------------------------------------------
extern "C" void kernel_launch(void* const* d_in, const int* in_sizes, int n_in,
                              void* d_out, int out_size, void* d_ws, size_t ws_size,
                              hipStream_t stream) {
  (void)in_sizes; (void)n_in; (void)out_size; (void)ws_size;
  const float* reps   = (const float*)d_in[0];
  const int*   qidx   = (const int*)d_in[1];
  const float* thrraw = (const float*)d_in[2];
  const float* strraw = (const float*)d_in[3];
  const float* wscale = (const float*)d_in[4];
  const float* temp   = (const float*)d_in[5];
  float* out = (float*)d_out;

  float* ws      = (float*)d_ws;
  float* qn_ws   = ws;                       // B*D
  float* acc_ws  = ws + Bn * Dn;             // B*D
  float* sims_ws = ws + 2 * Bn * Dn;         // B*R (reused as adjusted weights)
  float* flag_ws = ws + 2 * Bn * Dn + Bn * Rn;  // B

  prep_kernel<<<Bn, 256, 0, stream>>>(reps, qidx, qn_ws, acc_ws);
  sims_kernel<<<dim3(Rn / TILE_ROWS, Bn), 256, 0, stream>>>(reps, qidx, qn_ws, out, sims_ws);
  weights_kernel<<<Bn, 256, 0, stream>>>(sims_ws, thrraw, wscale, temp, flag_ws);
  wsum_kernel<<<dim3(Rn / 256, Bn), 256, 0, stream>>>(reps, sims_ws, acc_ws);
  finalize_kernel<<<Bn, 256, 0, stream>>>(reps, qidx, acc_ws, flag_ws, strraw, out);
}